// HeatEquationLoss_14998025798279
// MI455X (gfx1250) — compile-verified
//
#include <hip/hip_runtime.h>

#define ALPHA_C   0.0257f
#define SRC_I     100000.0f
#define THRESH    1000.0f
#define N_TOTAL   16777216.0f   // 8*128*128*128
#define NBLK      16384         // (8*128*128)/8 rows-per-block

typedef float v2f __attribute__((ext_vector_type(2)));
typedef float v8f __attribute__((ext_vector_type(8)));

// Wave-wide f32 sum using the f32 WMMA (exact in fp32):
// A = ones(16x4), B(4x16) carries one partial per lane (other slot 0).
// D[m,n] = sum_k B[k,n]  -> c[0] over lanes 0..15 enumerates all 16 column
// sums (lanes 16..31 duplicate them, row M=8), so 4 xor-shuffles give the
// full 32-lane total in every lane. Robust to any K/N layout permutation.
__device__ __forceinline__ float wave_sum_wmma(float x) {
    v2f a; a[0] = 1.0f; a[1] = 1.0f;
    v2f b; b[0] = x;    b[1] = 0.0f;
    v8f c = {};
    c = __builtin_amdgcn_wmma_f32_16x16x4_f32(
            /*neg_a=*/false, a, /*neg_b=*/false, b,
            /*c_mod=*/(short)0, c, /*reuse_a=*/false, /*reuse_b=*/false);
    float s = c[0];
    s += __shfl_xor(s, 1, 32);
    s += __shfl_xor(s, 2, 32);
    s += __shfl_xor(s, 4, 32);
    s += __shfl_xor(s, 8, 32);
    return s;
}

__global__ __launch_bounds__(256) void heat_residual_partial(
    const float* __restrict__ mod,   // model_output
    const float* __restrict__ cur,   // current_input
    float* __restrict__ partial) {
    const int lane = threadIdx.x & 31;
    const int wave = threadIdx.x >> 5;
    const int row  = blockIdx.x * 8 + wave;      // flat (b*128 + d)*128 + h
    const int h    = row & 127;
    const int d    = (row >> 7) & 127;
    const int idx  = (row << 7) + (lane << 2);   // element index, 16B aligned

    const float4 c4 = *(const float4*)(cur + idx);
    const float4 m4 = *(const float4*)(mod + idx);

    float4 hm = make_float4(0.f, 0.f, 0.f, 0.f);
    float4 hp = hm, dm = hm, dp = hm;
    if (h > 0)   hm = *(const float4*)(cur + idx - 128);
    if (h < 127) hp = *(const float4*)(cur + idx + 128);
    if (d > 0)   dm = *(const float4*)(cur + idx - 16384);
    if (d < 127) dp = *(const float4*)(cur + idx + 16384);
    if (d < 126) __builtin_prefetch(cur + idx + 32768, 0, 1); // d+2 slice

    // W-direction neighbors: register shifts + lane shuffles, zero padding.
    float left  = __shfl_up(c4.w, 1, 32);
    float right = __shfl_down(c4.x, 1, 32);
    left  = (lane == 0)  ? 0.0f : left;
    right = (lane == 31) ? 0.0f : right;

    const float cc[4] = {c4.x, c4.y, c4.z, c4.w};
    const float mm[4] = {m4.x, m4.y, m4.z, m4.w};
    const float wl[4] = {left, c4.x, c4.y, c4.z};
    const float wr[4] = {c4.y, c4.z, c4.w, right};
    const float hmv[4] = {hm.x, hm.y, hm.z, hm.w};
    const float hpv[4] = {hp.x, hp.y, hp.z, hp.w};
    const float dmv[4] = {dm.x, dm.y, dm.z, dm.w};
    const float dpv[4] = {dp.x, dp.y, dp.z, dp.w};

    float acc = 0.0f;
#pragma unroll
    for (int i = 0; i < 4; ++i) {
        float lap = wl[i] + wr[i] + hmv[i] + hpv[i] + dmv[i] + dpv[i]
                  - 6.0f * cc[i];
        float td  = (mm[i] - cc[i]) / 3.0f;              // DELTA_T = 3.0
        float src = (cc[i] > THRESH) ? SRC_I : 0.0f;
        float r   = td - ALPHA_C * lap - src;
        acc = fmaf(r, r, acc);
    }

    // EXEC is full here (only uniform branches above): safe for WMMA.
    float wsum = wave_sum_wmma(acc);

    __shared__ float s_wave[8];
    if (lane == 0) s_wave[wave] = wsum;
    __syncthreads();
    if (threadIdx.x == 0) {
        float s = 0.0f;
#pragma unroll
        for (int i = 0; i < 8; ++i) s += s_wave[i];
        partial[blockIdx.x] = s;
    }
}

__global__ __launch_bounds__(256) void heat_residual_reduce(
    const float* __restrict__ partial, float* __restrict__ out) {
    // NBLK partials, one block, float4 strided loads (16 iterations).
    const float4* p4 = (const float4*)partial;
    float acc = 0.0f;
#pragma unroll 4
    for (int i = threadIdx.x; i < NBLK / 4; i += 256) {
        float4 v = p4[i];
        acc += (v.x + v.y) + (v.z + v.w);
    }
    float wsum = wave_sum_wmma(acc);

    __shared__ float s_wave[8];
    const int lane = threadIdx.x & 31;
    const int wave = threadIdx.x >> 5;
    if (lane == 0) s_wave[wave] = wsum;
    __syncthreads();
    if (threadIdx.x == 0) {
        float s = 0.0f;
#pragma unroll
        for (int i = 0; i < 8; ++i) s += s_wave[i];
        out[0] = s / N_TOTAL;
    }
}

extern "C" void kernel_launch(void* const* d_in, const int* in_sizes, int n_in,
                              void* d_out, int out_size, void* d_ws, size_t ws_size,
                              hipStream_t stream) {
    const float* mod = (const float*)d_in[0];   // model_output
    const float* cur = (const float*)d_in[1];   // current_input
    float* out       = (float*)d_out;
    float* partial   = (float*)d_ws;            // NBLK floats = 64 KB

    heat_residual_partial<<<NBLK, 256, 0, stream>>>(mod, cur, partial);
    heat_residual_reduce<<<1, 256, 0, stream>>>(partial, out);
}